// OTDetLoss_57964878627380
// MI455X (gfx1250) — compile-verified
//
#include <hip/hip_runtime.h>
#include <hip/hip_bf16.h>

typedef __attribute__((ext_vector_type(2))) float v2f;
typedef __attribute__((ext_vector_type(8))) float v8f;

#define NCOMP 6
// component slots: 0=cls focal, 1=reg smoothL1, 2=angle, 3=(1-iou), 4=obj bce, 5=fg count
#define NCLS 15
#define PART_STRIDE 8   // padded stride in workspace for alignment

__device__ __forceinline__ float smooth_l1(float x) {
    float d = fabsf(x);
    return (d < 1.0f) ? 0.5f * d * d : d - 0.5f;
}

__global__ __launch_bounds__(256) void otdet_stage1(
    const float* __restrict__ centers,     // [N,2]
    const float* __restrict__ wh,          // [N,2]
    const float* __restrict__ angles,      // [N,1]
    const float* __restrict__ cls_logits,  // [N,15]
    const float* __restrict__ conf,        // [N,1]
    const float* __restrict__ tgt,         // [N,5]
    const int*   __restrict__ labels,      // [N]
    const int*   __restrict__ img_size,    // [1]
    float*       __restrict__ part,        // [nblk*PART_STRIDE]
    int N)
{
    const int s = blockIdx.x * 256 + threadIdx.x;
    float acc[NCOMP] = {0.f, 0.f, 0.f, 0.f, 0.f, 0.f};

    if (s < N) {
        const float inv_s = 1.0f / (float)(*img_size);
        const int lab = labels[s];
        const float fm = (lab >= 0) ? 1.0f : 0.0f;   // == fg_mask by construction

        // ---- classification focal loss over 15 classes ----
        float cls = 0.f;
        #pragma unroll
        for (int c = 0; c < NCLS; ++c) {
            float x  = cls_logits[s * NCLS + c];
            float oh = (c == lab) ? 1.0f : 0.0f;
            float p  = 1.0f / (1.0f + expf(-x));
            float bce = fmaxf(x, 0.0f) - x * oh + log1pf(expf(-fabsf(x)));
            float p_t = p * oh + (1.0f - p) * (1.0f - oh);
            float a_t = 0.25f * oh + 0.75f * (1.0f - oh);
            float om  = 1.0f - p_t;
            cls += a_t * om * om * bce;
        }
        acc[0] = cls;

        // ---- box regression (smooth L1 on normalized coords) ----
        const float cx = centers[2 * s + 0], cy = centers[2 * s + 1];
        const float pw = wh[2 * s + 0],      ph = wh[2 * s + 1];
        const float g0 = tgt[5 * s + 0], g1 = tgt[5 * s + 1];
        const float g2 = tgt[5 * s + 2], g3 = tgt[5 * s + 3];
        const float ga = tgt[5 * s + 4];
        float reg = smooth_l1((cx - g0) * inv_s) + smooth_l1((cy - g1) * inv_s)
                  + smooth_l1((pw - g2) * inv_s) + smooth_l1((ph - g3) * inv_s);
        acc[1] = reg * fm;

        // ---- angle loss (double-angle sin/cos smooth L1) ----
        const float pa2 = 2.0f * angles[s];
        const float ga2 = 2.0f * ga;
        float ang = smooth_l1(sinf(pa2) - sinf(ga2)) + smooth_l1(cosf(pa2) - cosf(ga2));
        acc[2] = ang * fm;

        // ---- aligned IoU loss ----
        const float p1x = cx - pw * 0.5f, p1y = cy - ph * 0.5f;
        const float p2x = cx + pw * 0.5f, p2y = cy + ph * 0.5f;
        const float q1x = g0 - g2 * 0.5f, q1y = g1 - g3 * 0.5f;
        const float q2x = g0 + g2 * 0.5f, q2y = g1 + g3 * 0.5f;
        const float ix = fmaxf(fminf(p2x, q2x) - fmaxf(p1x, q1x), 0.0f);
        const float iy = fmaxf(fminf(p2y, q2y) - fmaxf(p1y, q1y), 0.0f);
        const float inter = ix * iy;
        const float iou = inter / (pw * ph + g2 * g3 - inter + 1e-7f);
        acc[3] = (1.0f - iou) * fm;

        // ---- objectness BCE ----
        const float cf  = conf[s];
        const float lc  = fmaxf(logf(cf), -100.0f);
        const float l1c = fmaxf(log1pf(-cf), -100.0f);
        acc[4] = -(fm * lc + (1.0f - fm) * l1c);

        acc[5] = fm;
    }

    // ---- deterministic block reduction: wave shuffle -> LDS -> single writer ----
    __shared__ float smem[8][NCOMP];
    const int lane = threadIdx.x & 31;
    const int wave = threadIdx.x >> 5;
    #pragma unroll
    for (int j = 0; j < NCOMP; ++j) {
        float v = acc[j];
        #pragma unroll
        for (int off = 16; off > 0; off >>= 1)
            v += __shfl_down(v, off, 32);
        if (lane == 0) smem[wave][j] = v;
    }
    __syncthreads();
    if (threadIdx.x == 0) {
        #pragma unroll
        for (int j = 0; j < NCOMP; ++j) {
            float t = 0.f;
            #pragma unroll
            for (int w = 0; w < 8; ++w) t += smem[w][j];
            part[blockIdx.x * PART_STRIDE + j] = t;
        }
    }
}

// Single-wave epilogue: deterministic lane-strided accumulation of per-block
// partials, then an exact f32 cross-lane sum via V_WMMA_F32_16X16X4_F32 with
// B = all-ones (layout-independent). Per the documented A(16x4) layout,
// rowsum(m) = x_m + x_{m+16}; summing the 8 D VGPRs at lanes 0 and 16 gives
// lanes {0..7,16..23} and {8..15,24..31} respectively.
__global__ __launch_bounds__(32) void otdet_stage2(
    const float* __restrict__ part, int nblk,
    float* __restrict__ out, int BK)
{
    const int lane = threadIdx.x;
    float acc[NCOMP] = {0.f, 0.f, 0.f, 0.f, 0.f, 0.f};
    for (int b = lane; b < nblk; b += 32) {
        #pragma unroll
        for (int j = 0; j < NCOMP; ++j)
            acc[j] += part[b * PART_STRIDE + j];
    }

    float tot[NCOMP];
#if __has_builtin(__builtin_amdgcn_wmma_f32_16x16x4_f32)
    v2f bones; bones.x = 1.0f; bones.y = 1.0f;
    #pragma unroll
    for (int j = 0; j < NCOMP; ++j) {
        v2f a; a.x = acc[j]; a.y = 0.0f;
        v8f cz = {};
        v8f d = __builtin_amdgcn_wmma_f32_16x16x4_f32(
            false, a, false, bones, (short)0, cz, false, false);
        float t = d[0] + d[1] + d[2] + d[3] + d[4] + d[5] + d[6] + d[7];
        tot[j] = __shfl(t, 0, 32) + __shfl(t, 16, 32);
    }
#else
    #pragma unroll
    for (int j = 0; j < NCOMP; ++j) {
        float v = acc[j];
        #pragma unroll
        for (int off = 16; off > 0; off >>= 1)
            v += __shfl_down(v, off, 32);
        tot[j] = __shfl(v, 0, 32);
    }
#endif

    if (lane == 0) {
        const float n_fg = fmaxf(tot[5], 1.0f);   // == n_fg_cls == num_fg
        // CLS_W=1, REG_W=5, ANG_W=1, IOU_W=2, OT_W=1
        float total = 1.0f * tot[0] / n_fg
                    + 5.0f * tot[1] / n_fg
                    + 1.0f * tot[2] / n_fg
                    + 2.0f * tot[3] / n_fg
                    + 1.0f * tot[4] / (float)BK;
        out[0] = total;
    }
}

extern "C" void kernel_launch(void* const* d_in, const int* in_sizes, int n_in,
                              void* d_out, int out_size, void* d_ws, size_t ws_size,
                              hipStream_t stream) {
    const float* centers    = (const float*)d_in[0];
    const float* wh         = (const float*)d_in[1];
    const float* angles     = (const float*)d_in[2];
    const float* cls_logits = (const float*)d_in[3];
    const float* conf       = (const float*)d_in[4];
    const float* tgt        = (const float*)d_in[5];
    const int*   labels     = (const int*)d_in[6];
    // d_in[7] = fg_mask (unused; identical to labels >= 0 by construction)
    const int*   img_size   = (const int*)d_in[8];

    const int N    = in_sizes[0] / 2;          // centers has N*2 elements
    const int nblk = (N + 255) / 256;          // 2100 for B=64, K=8400
    float* part = (float*)d_ws;                // nblk * PART_STRIDE floats

    otdet_stage1<<<nblk, 256, 0, stream>>>(centers, wh, angles, cls_logits,
                                           conf, tgt, labels, img_size,
                                           part, N);
    otdet_stage2<<<1, 32, 0, stream>>>(part, nblk, (float*)d_out, N);
}